// TripletLoss_out_69672959475956
// MI455X (gfx1250) — compile-verified
//
#include <hip/hip_runtime.h>
#include <hip/hip_bf16.h>

// CDNA5 / gfx1250, wave32.
typedef __attribute__((ext_vector_type(2))) float v2f;
typedef __attribute__((ext_vector_type(8))) float v8f;
typedef __attribute__((ext_vector_type(4))) int   v4i;

#define MARGIN 0.3f
#define NMAX   16384   // B array fully LDS-resident (64 KB of 320 KB/WGP)

#if __has_builtin(__builtin_amdgcn_global_load_async_to_lds_b128)
#define HAS_ASYNC_LDS 1
typedef __attribute__((address_space(1))) v4i gas_v4i;   // global int4
typedef __attribute__((address_space(3))) v4i lds_v4i;   // LDS int4
#endif

// ---------------------------------------------------------------------------
// Kernel 1: s = sigmoid(x); fold masks into sentinels so the pair kernel is
// branch-free:  A[i] = pos ? s : 2.0  (hinge(0.3-2.0+b)=0 for all b<1)
//               B[j] = neg ? s : -10  (hinge(0.3-a-10)=0 for all a>0)
// ---------------------------------------------------------------------------
__global__ __launch_bounds__(256) void prep_kernel(const float* __restrict__ x,
                                                   const int* __restrict__ tg,
                                                   float* __restrict__ A,
                                                   float* __restrict__ B, int n) {
    int i = blockIdx.x * 256 + threadIdx.x;
    if (i < n) {
        float s = 1.0f / (1.0f + __expf(-x[i]));
        int t = tg[i];
        A[i] = (t == 1) ? s : 2.0f;
        B[i] = (t == 0) ? s : -10.0f;
    }
}

// ---------------------------------------------------------------------------
// Kernel 2: all-pairs hinge with WMMA f32 16x16x4 as the exact accumulator.
// Whole B array staged once into LDS via GLOBAL_LOAD_ASYNC_TO_LDS_B128
// (ASYNCcnt path), then each wave owns 16 rows and per WMMA step each lane
// computes two hinge values (2 VALU ops, packed); the matrix core folds the
// 16x4 block into the persistent 16x16 f32 C accumulator via a ones operand:
//   D[m,n] = sum_k Ahinge[m,k] * 1 + C[m,n]
// ---------------------------------------------------------------------------
__global__ __launch_bounds__(256) void pairs_kernel(const float* __restrict__ A,
                                                    const float* __restrict__ B,
                                                    float* __restrict__ partial,
                                                    int n) {
    __shared__ __align__(16) float bsh[NMAX];   // 64 KB: entire B array
    __shared__ float wsum[8];

    const int tid  = threadIdx.x;
    const int lane = tid & 31;
    const int wave = tid >> 5;
    // 8 waves * 16 rows = 128 rows per block
    const int row  = blockIdx.x * 128 + wave * 16 + (lane & 15);

    const float a   = A[row];            // lanes L and L^16 share the same row value
    const float cma = MARGIN - a;

    // ---- stage all of B into LDS (async-to-LDS if available) ----
#if defined(HAS_ASYNC_LDS)
    {
        for (int t = tid; t < n / 4; t += 256) {
            __builtin_amdgcn_global_load_async_to_lds_b128(
                (gas_v4i*)(B + 4 * t), (lds_v4i*)&bsh[4 * t],
                /*offset=*/0, /*cpol=*/0);
        }
#if __has_builtin(__builtin_amdgcn_s_wait_asynccnt)
        __builtin_amdgcn_s_wait_asynccnt(0);
#else
        asm volatile("s_wait_asynccnt 0" ::: "memory");
#endif
    }
#else
    {
        const float4* src = reinterpret_cast<const float4*>(B);
        float4* dst = reinterpret_cast<float4*>(bsh);
        for (int t = tid; t < n / 4; t += 256) dst[t] = src[t];
    }
#endif
    __syncthreads();

    v8f acc = {};                        // 16x16 f32 C/D accumulator (8 VGPRs)
    const v2f ones = {1.0f, 1.0f};       // B operand: all ones -> row-sum
    const int koff = (lane >> 4) << 1;   // lanes 0-15: k 0,1 ; lanes 16-31: k 2,3

#pragma unroll 8
    for (int k = 0; k < n; k += 4) {
        const float b0 = bsh[k + koff];
        const float b1 = bsh[k + koff + 1];
        v2f av;
        av.x = fmaxf(cma + b0, 0.0f);
        av.y = fmaxf(cma + b1, 0.0f);
        // D = Ahinge(16x4) * ones(4x16) + C : exact f32 row-sum accumulate
        acc = __builtin_amdgcn_wmma_f32_16x16x4_f32(
            false, av, false, ones, (short)0, acc, false, false);
    }

    // C[m,n] is identical across columns n. Lanes 0-15 all hold rows 0-7,
    // lanes 16-31 hold rows 8-15: one vertical add + shfl_xor(16) finishes.
    float t8 = acc[0] + acc[1] + acc[2] + acc[3] +
               acc[4] + acc[5] + acc[6] + acc[7];
    float wtot = t8 + __shfl_xor(t8, 16, 32);   // full 16-row sum, all lanes

    if (lane == 0) wsum[wave] = wtot;
    __syncthreads();
    if (tid == 0) {
        float s = 0.0f;
        for (int w = 0; w < 8; ++w) s += wsum[w];
        partial[blockIdx.x] = s;                 // deterministic: fixed order
    }
}

// ---------------------------------------------------------------------------
// Kernel 3: deterministic final reduction of per-block partials + pos count.
// ---------------------------------------------------------------------------
__global__ __launch_bounds__(256) void finalize_kernel(const float* __restrict__ partial,
                                                       int nblocks,
                                                       const int* __restrict__ tg,
                                                       int n,
                                                       float* __restrict__ out) {
    __shared__ float fs[256];
    __shared__ int   pc[256];
    int tid = threadIdx.x;

    float s = 0.0f;
    for (int i = tid; i < nblocks; i += 256) s += partial[i];
    int p = 0;
    for (int i = tid; i < n; i += 256) p += (tg[i] == 1) ? 1 : 0;
    fs[tid] = s;
    pc[tid] = p;
    __syncthreads();
    for (int off = 128; off > 0; off >>= 1) {
        if (tid < off) { fs[tid] += fs[tid + off]; pc[tid] += pc[tid + off]; }
        __syncthreads();
    }
    if (tid == 0) {
        float P = (float)pc[0];
        float Q = (float)(n - pc[0]);
        out[0] = fs[0] / (P * Q);
    }
}

// ---------------------------------------------------------------------------
extern "C" void kernel_launch(void* const* d_in, const int* in_sizes, int n_in,
                              void* d_out, int out_size, void* d_ws, size_t ws_size,
                              hipStream_t stream) {
    const float* x  = (const float*)d_in[0];
    const int*   tg = (const int*)d_in[1];
    const int n = in_sizes[0];            // 16384

    float* ws      = (float*)d_ws;
    float* A       = ws;                  // n floats
    float* B       = ws + n;              // n floats
    float* partial = ws + 2 * n;          // n/128 floats
    float* out     = (float*)d_out;

    prep_kernel<<<(n + 255) / 256, 256, 0, stream>>>(x, tg, A, B, n);

    const int nblk = n / 128;             // 128 blocks of 8 waves x 16 rows
    pairs_kernel<<<nblk, 256, 0, stream>>>(A, B, partial, n);

    finalize_kernel<<<1, 256, 0, stream>>>(partial, nblk, tg, n, out);
}